// DeNuCLoss_87170656239837
// MI455X (gfx1250) — compile-verified
//
#include <hip/hip_runtime.h>
#include <hip/hip_bf16.h>

// Problem constants (fixed by the reference)
#define BS 8
#define NQ 16384
#define NG 1024
#define TOPK 4

typedef float v2f __attribute__((ext_vector_type(2)));
typedef float v8f __attribute__((ext_vector_type(8)));

// Monotone float -> u32 encoding (ascending float order == ascending uint order)
__device__ __forceinline__ unsigned int fenc(float f) {
    unsigned int u = __float_as_uint(f);
    return u ^ ((unsigned int)((int)u >> 31) | 0x80000000u);
}

// Branchless sorted insert into ascending top-4 (t0 <= t1 <= t2 <= t3)
__device__ __forceinline__ void insert4(unsigned long long k,
                                        unsigned long long& t0, unsigned long long& t1,
                                        unsigned long long& t2, unsigned long long& t3) {
    bool b3 = k < t3;
    bool b2 = k < t2;
    bool b1 = k < t1;
    bool b0 = k < t0;
    unsigned long long n3 = b2 ? t2 : (b3 ? k : t3);
    unsigned long long n2 = b1 ? t1 : (b2 ? k : t2);
    unsigned long long n1 = b0 ? t0 : (b1 ? k : t1);
    unsigned long long n0 = b0 ? k : t0;
    t3 = n3; t2 = n2; t1 = n1; t0 = n0;
}

// ---------------- Kernel 1: init workspace ----------------
__global__ __launch_bounds__(256) void init_kernel(unsigned int* __restrict__ flags,
                                                   float* __restrict__ accum) {
    int i = blockIdx.x * 256 + threadIdx.x;
    if (i < BS * NQ) flags[i] = 0u;
    if (i == 0) { accum[0] = 0.0f; accum[1] = 0.0f; }
}

// ---------------- Kernel 2: per-query pack {x, y, -p0, -p1} ----------------
__global__ __launch_bounds__(256) void qpack_kernel(const float* __restrict__ pred_coords,
                                                    const float* __restrict__ pred_logits,
                                                    float4* __restrict__ qpack) {
    int i = blockIdx.x * 256 + threadIdx.x;
    if (i >= BS * NQ) return;
    float x  = pred_coords[2 * i + 0];
    float y  = pred_coords[2 * i + 1];
    float l0 = pred_logits[2 * i + 0];
    float l1 = pred_logits[2 * i + 1];
    float m  = fmaxf(l0, l1);
    float e0 = __expf(l0 - m);
    float e1 = __expf(l1 - m);
    float inv = 1.0f / (e0 + e1);
    qpack[i] = make_float4(x, y, -e0 * inv, -e1 * inv);
}

// ---------------- Kernel 3: WMMA cost tiles + streaming top-4 per gt column ----------------
// grid: BS * (NG/16) blocks; block: 256 threads (8 waves).
// Block (b, gtile): all 8 waves share the same 16 gt columns, each wave streams
// q-tiles t = wave, wave+8, ... (128 tiles of 16 queries).
__global__ __launch_bounds__(256) void cost_topk_kernel(const float* __restrict__ gt_coords,
                                                        const int* __restrict__ gt_labels,
                                                        const int* __restrict__ gt_masks,
                                                        const float4* __restrict__ qpack,
                                                        unsigned int* __restrict__ flags,
                                                        float* __restrict__ accum) {
    __shared__ unsigned long long smem[8][32][4];

    const int blk   = blockIdx.x;
    const int b     = blk >> 6;        // / (NG/16)
    const int gtile = blk & 63;
    const int tid   = threadIdx.x;
    const int lane  = tid & 31;
    const int wave  = tid >> 5;
    const int col   = lane & 15;       // gt column within the 16-wide tile
    const bool hi   = (lane >= 16);    // lanes 16-31 carry K=2,3 of the fragments

    // ---- loop-invariant B fragments (4x16, K x N) ----
    const int g = gtile * 16 + col;
    const float gx = gt_coords[(b * NG + g) * 2 + 0];
    const float gy = gt_coords[(b * NG + g) * 2 + 1];
    const int  lbl = gt_labels[b * NG + g];

    // B1 rows: [-2gx, -2gy, 1, |g|^2]  -> distance^2 via one WMMA
    v2f bf1;
    bf1.x = hi ? 1.0f                : (-2.0f * gx);
    bf1.y = hi ? (gx * gx + gy * gy) : (-2.0f * gy);
    // B2 rows: one-hot(label) -> picks -p_label from A2
    v2f bf2;
    bf2.x = hi ? 0.0f : (lbl == 0 ? 1.0f : 0.0f);
    bf2.y = hi ? 0.0f : (lbl == 1 ? 1.0f : 0.0f);

    // Two independent top-4 accumulators (r=0..3 and r=4..7) to shorten the
    // serial v_min_u64 dependency chain; merged after the loop.
    unsigned long long a0 = ~0ull, a1 = ~0ull, a2 = ~0ull, a3 = ~0ull;
    unsigned long long c0 = ~0ull, c1 = ~0ull, c2 = ~0ull, c3 = ~0ull;
    const float4* __restrict__ qb = qpack + b * NQ;

    for (int t = wave; t < NQ / 16; t += 8) {
        const int qbase = t * 16;
        const float4 qp = qb[qbase + col];   // {x, y, -p0, -p1} for query qbase+col

        // Prefetch this wave's next q-tile (clamped; branch-free so EXEC stays all-1s)
        int pq = qbase + 128 + col;
        pq = pq < NQ ? pq : (qbase + col);
        __builtin_prefetch(qb + pq, 0, 1);

        // A1 rows: [x, y, |p|^2, 1] ; A2 rows: [-p0, -p1, 0, 0]
        v2f fa1, fa2;
        fa1.x = hi ? (qp.x * qp.x + qp.y * qp.y) : qp.x;
        fa1.y = hi ? 1.0f : qp.y;
        fa2.x = hi ? 0.0f : qp.z;
        fa2.y = hi ? 0.0f : qp.w;

        v8f d1 = {};
        v8f d2 = {};
        // d1[m][n] = |p_m|^2 + |g_n|^2 - 2 p_m.g_n   (squared distance)
        d1 = __builtin_amdgcn_wmma_f32_16x16x4_f32(false, fa1, false, bf1, (short)0, d1, false, false);
        // d2[m][n] = -prob[m][label_n]
        d2 = __builtin_amdgcn_wmma_f32_16x16x4_f32(false, fa2, false, bf2, (short)0, d2, false, false);

        const int qoff = qbase + (hi ? 8 : 0);
#pragma unroll
        for (int r = 0; r < 8; ++r) {
            float s = d1[r];
            s = s > 0.0f ? s : 0.0f;
            // Raw v_sqrt_f32 (~1 ulp): plenty for top-k ordering of random costs,
            // avoids the branchy precise-sqrt expansion inside the WMMA loop.
            float cost = 0.1f * __builtin_amdgcn_sqrtf(s) + d2[r];
            // D layout: vgpr r, lanes 0-15 -> M=r ; lanes 16-31 -> M=r+8 ; N = lane&15
            unsigned long long key =
                ((unsigned long long)fenc(cost) << 32) | (unsigned int)(qoff + r);
            if (r < 4) insert4(key, a0, a1, a2, a3);
            else       insert4(key, c0, c1, c2, c3);
        }
    }

    // Merge accumulator B into A (full u64 keys -> exact (cost, index) tie-break)
    insert4(c0, a0, a1, a2, a3);
    insert4(c1, a0, a1, a2, a3);
    insert4(c2, a0, a1, a2, a3);
    insert4(c3, a0, a1, a2, a3);

    smem[wave][lane][0] = a0;
    smem[wave][lane][1] = a1;
    smem[wave][lane][2] = a2;
    smem[wave][lane][3] = a3;
    __syncthreads();

    // ---- merge: column n's candidates live at lanes n and n+16 of every wave ----
    if (tid < 16) {
        unsigned long long m0 = ~0ull, m1 = ~0ull, m2 = ~0ull, m3 = ~0ull;
#pragma unroll
        for (int w = 0; w < 8; ++w) {
#pragma unroll
            for (int k = 0; k < 4; ++k) insert4(smem[w][tid][k],      m0, m1, m2, m3);
#pragma unroll
            for (int k = 0; k < 4; ++k) insert4(smem[w][tid + 16][k], m0, m1, m2, m3);
        }
        const int gg = gtile * 16 + tid;
        if (gt_masks[b * NG + gg] != 0) {   // masked columns are irrelevant downstream
            const float ggx = gt_coords[(b * NG + gg) * 2 + 0];
            const float ggy = gt_coords[(b * NG + gg) * 2 + 1];
            unsigned long long keys[4] = { m0, m1, m2, m3 };
            float local = 0.0f;
#pragma unroll
            for (int k = 0; k < TOPK; ++k) {
                unsigned int q = (unsigned int)(keys[k] & 0xffffffffu);
                float4 qp = qb[q];
                float dx = qp.x - ggx;
                float dy = qp.y - ggy;
                local += dx * dx + dy * dy;
                flags[b * NQ + q] = 1u;      // benign race: everyone writes 1
            }
            atomicAdd(&accum[0], local);
        }
    }
}

// ---------------- Kernel 4: classification NLL sum ----------------
__global__ __launch_bounds__(256) void cls_kernel(const float* __restrict__ pred_logits,
                                                  const unsigned int* __restrict__ flags,
                                                  float* __restrict__ accum) {
    __shared__ float red[256];
    int i = blockIdx.x * 256 + threadIdx.x;
    float v = 0.0f;
    if (i < BS * NQ) {
        float l0 = pred_logits[2 * i + 0];
        float l1 = pred_logits[2 * i + 1];
        float m  = fmaxf(l0, l1);
        float lse = m + __logf(__expf(l0 - m) + __expf(l1 - m));
        float chosen = flags[i] ? l0 : l1;   // label 0 if matched, else N_CLS(=1)
        v = chosen - lse;
    }
    red[threadIdx.x] = v;
    __syncthreads();
#pragma unroll
    for (int s = 128; s > 0; s >>= 1) {
        if (threadIdx.x < s) red[threadIdx.x] += red[threadIdx.x + s];
        __syncthreads();
    }
    if (threadIdx.x == 0) atomicAdd(&accum[1], red[0]);
}

// ---------------- Kernel 5: finalize ----------------
__global__ __launch_bounds__(256) void finalize_kernel(const int* __restrict__ gt_masks,
                                                       const float* __restrict__ accum,
                                                       float* __restrict__ out) {
    __shared__ int red[256];
    int cnt = 0;
    for (int i = threadIdx.x; i < BS * NG; i += 256) cnt += (gt_masks[i] != 0);
    red[threadIdx.x] = cnt;
    __syncthreads();
#pragma unroll
    for (int s = 128; s > 0; s >>= 1) {
        if (threadIdx.x < s) red[threadIdx.x] += red[threadIdx.x + s];
        __syncthreads();
    }
    if (threadIdx.x == 0) {
        float n_valid = (float)red[0] * (float)TOPK;             // sum(masks) * TOP_K
        out[0] = accum[0] / (n_valid * 2.0f) * 5.0f;             // REG_COEF
        out[1] = -accum[1] / (float)(BS * NQ);                   // CLS_COEF = 1
    }
}

extern "C" void kernel_launch(void* const* d_in, const int* in_sizes, int n_in,
                              void* d_out, int out_size, void* d_ws, size_t ws_size,
                              hipStream_t stream) {
    const float* pred_coords = (const float*)d_in[0];  // (8,16384,2)
    const float* pred_logits = (const float*)d_in[1];  // (8,16384,2)
    const float* gt_coords   = (const float*)d_in[2];  // (8,1024,2)
    const int*   gt_labels   = (const int*)d_in[3];    // (8,1024)
    const int*   gt_masks    = (const int*)d_in[4];    // (8,1024) nonzero == valid
    float* out = (float*)d_out;                        // [reg_loss, cls_loss]

    // Workspace layout: [accum 16B] [flags BS*NQ u32] [qpack BS*NQ float4]
    char* ws = (char*)d_ws;
    float*        accum = (float*)ws;
    unsigned int* flags = (unsigned int*)(ws + 16);
    float4*       qpack = (float4*)(ws + 16 + (size_t)BS * NQ * sizeof(unsigned int));

    const int n_bq = BS * NQ;                 // 131072
    const int blks_bq = (n_bq + 255) / 256;   // 512

    init_kernel<<<blks_bq, 256, 0, stream>>>(flags, accum);
    qpack_kernel<<<blks_bq, 256, 0, stream>>>(pred_coords, pred_logits, qpack);
    cost_topk_kernel<<<BS * (NG / 16), 256, 0, stream>>>(gt_coords, gt_labels, gt_masks,
                                                         qpack, flags, accum);
    cls_kernel<<<blks_bq, 256, 0, stream>>>(pred_logits, flags, accum);
    finalize_kernel<<<1, 256, 0, stream>>>(gt_masks, accum, out);
}